// HyperMod_9156870275528
// MI455X (gfx1250) — compile-verified
//
#include <hip/hip_runtime.h>

// HyperMod on MI455X (gfx1250, wave32).
// Memory-bound op (~2.5 GB moved, ~110us HBM floor; scatter destinations are
// L2-resident in the 192MB L2). GEMMs are tiny (5 GFLOP) -> keep exact fp32
// via V_WMMA_F32_16X16X4_F32; A-tiles staged with GLOBAL_LOAD_ASYNC_TO_LDS.

typedef __attribute__((ext_vector_type(2))) float v2f;
typedef __attribute__((ext_vector_type(8))) float v8f;

#define D_ 128
#define TM 32
#define LDS_PITCH 132  // 132 % 64 = 4 -> fragment reads hit 16 distinct banks

// Y[r, :] = relu(X[r, :] @ W + bias) * cscale * (HAS_RS ? rowscale[r] : 1)
// Block: 256 thr = 8 waves; handles TM=32 rows x 128 cols.
// Wave w owns column tile [16w, 16w+16) and both 16-row tiles (B frag reuse).
template <bool HAS_RS>
__global__ __launch_bounds__(256)
void gemm_relu_rowscale(const float* __restrict__ X,
                        const float* __restrict__ W,
                        const float* __restrict__ bias,
                        const float* __restrict__ rowscale,
                        float cscale,
                        float* __restrict__ Y,
                        int R)
{
    __shared__ float As[TM * LDS_PITCH];
    const int tid  = threadIdx.x;
    const int wave = tid >> 5;
    const int lane = tid & 31;
    const int half = lane >> 4;   // lanes 16-31 carry the K+2 pair
    const int r16  = lane & 15;
    const int row0 = blockIdx.x * TM;
    const bool full_tile = (row0 + TM <= R);   // uniform across block

    if (full_tile) {
        // CDNA5 async copy: memory -> LDS directly (no VGPR round trip),
        // 16B per lane per issue, tracked by ASYNCcnt.
        const unsigned lds_base = (unsigned)(uintptr_t)(&As[0]);
#pragma unroll
        for (int i = 0; i < 4; ++i) {
            int chunk = tid + i * 256;          // float4 chunk id 0..1023
            int rr = chunk >> 5;                // 32 chunks per 128-float row
            int cc = (chunk & 31) * 4;
            unsigned lds_addr = lds_base + (unsigned)(rr * LDS_PITCH + cc) * 4u;
            const float* gp = X + (size_t)(row0 + rr) * D_ + cc;
            asm volatile("global_load_async_to_lds_b128 %0, %1, off"
                         :
                         : "v"(lds_addr), "v"(gp)
                         : "memory");
        }
        asm volatile("s_wait_asynccnt 0x0" ::: "memory");
    } else {
        // Tail block (rows >= R must be zero-filled): VGPR staging path.
#pragma unroll
        for (int i = 0; i < 4; ++i) {
            int chunk = tid + i * 256;
            int rr = chunk >> 5;
            int cc = (chunk & 31) * 4;
            float4 val = make_float4(0.f, 0.f, 0.f, 0.f);
            if (row0 + rr < R)
                val = *(const float4*)(X + (size_t)(row0 + rr) * D_ + cc);
            *(float4*)(&As[rr * LDS_PITCH + cc]) = val;
        }
    }
    __syncthreads();

    const int col = wave * 16 + r16;     // output column this lane owns (C/D layout)
    v8f acc0 = {};                       // rows row0    .. row0+15
    v8f acc1 = {};                       // rows row0+16 .. row0+31

    // K = 128 in 32 steps of V_WMMA_F32_16X16X4_F32.
    // A 16x4 f32 frag: lane holds M=lane%16, K pair {k0, k0+1} with
    // k0 = 4*ks + 2*(lane/16).  B 4x16 frag mirrors: row K in VGPR, N=lane%16.
#pragma unroll 4
    for (int ks = 0; ks < 32; ++ks) {
        const int k0 = ks * 4 + 2 * half;
        v2f a0 = *(const v2f*)(&As[r16 * LDS_PITCH + k0]);
        v2f a1 = *(const v2f*)(&As[(16 + r16) * LDS_PITCH + k0]);
        v2f b;
        b.x = W[(size_t)k0 * D_ + col];
        b.y = W[(size_t)(k0 + 1) * D_ + col];
        acc0 = __builtin_amdgcn_wmma_f32_16x16x4_f32(
            false, a0, false, b, (short)0, acc0, false, false);
        acc1 = __builtin_amdgcn_wmma_f32_16x16x4_f32(
            false, a1, false, b, (short)0, acc1, false, false);
    }

    // Epilogue: C/D layout -> VGPR i holds row (i + 8*half), fixed column.
    const float colb = bias[col];
    if (full_tile) {
#pragma unroll
        for (int i = 0; i < 8; ++i) {
            const int m0 = row0 + i + 8 * half;
            const int m1 = m0 + 16;
            float t0 = fmaxf(acc0[i] + colb, 0.f) * cscale;
            float t1 = fmaxf(acc1[i] + colb, 0.f) * cscale;
            if (HAS_RS) { t0 *= rowscale[m0]; t1 *= rowscale[m1]; }
            Y[(size_t)m0 * D_ + col] = t0;
            Y[(size_t)m1 * D_ + col] = t1;
        }
    } else {
#pragma unroll
        for (int i = 0; i < 8; ++i) {
            const int m0 = row0 + i + 8 * half;
            const int m1 = m0 + 16;
            float t0 = fmaxf(acc0[i] + colb, 0.f) * cscale;
            float t1 = fmaxf(acc1[i] + colb, 0.f) * cscale;
            if (m0 < R) {
                if (HAS_RS) t0 *= rowscale[m0];
                Y[(size_t)m0 * D_ + col] = t0;
            }
            if (m1 < R) {
                if (HAS_RS) t1 *= rowscale[m1];
                Y[(size_t)m1 * D_ + col] = t1;
            }
        }
    }
}

// One wave per incidence: dst[sidx[m], :] += src[gidx[m], :] * w[m]
// 32 lanes x float4 covers the 128-float row; atomics resolve in L2.
__global__ __launch_bounds__(256)
void scatter_rows_atomic(const float* __restrict__ src,
                         const int* __restrict__ gidx,
                         const int* __restrict__ sidx,
                         const float* __restrict__ w,
                         float* __restrict__ dst,
                         int m_total)
{
    const int wid  = (int)((blockIdx.x * (unsigned)blockDim.x + threadIdx.x) >> 5);
    const int lane = threadIdx.x & 31;
    if (wid >= m_total) return;
    const int g = gidx[wid];
    const int s = sidx[wid];
    const float ww = w[wid];
    const float4 val = *(const float4*)(src + (size_t)g * D_ + lane * 4);
    float* dp = dst + (size_t)s * D_ + lane * 4;
    atomicAdd(dp + 0, val.x * ww);
    atomicAdd(dp + 1, val.y * ww);
    atomicAdd(dp + 2, val.z * ww);
    atomicAdd(dp + 3, val.w * ww);
}

// y[i] = x[i] * cmul * (rs ? rs[row(i)] : 1), float4-vectorized (D=128 rows).
__global__ __launch_bounds__(256)
void rowscale_copy(const float* __restrict__ x,
                   const float* __restrict__ rs,
                   float cmul,
                   float* __restrict__ y,
                   long n4)
{
    long i = (long)blockIdx.x * 256 + threadIdx.x;
    if (i >= n4) return;
    float4 v = ((const float4*)x)[i];
    float s = cmul;
    if (rs) s *= rs[i >> 5];   // 32 float4 per 128-float row
    v.x *= s; v.y *= s; v.z *= s; v.w *= s;
    ((float4*)y)[i] = v;
}

// y[i] /= rs[row(i)] in place.
__global__ __launch_bounds__(256)
void rowdiv_inplace(float* __restrict__ y, const float* __restrict__ rs, long n4)
{
    long i = (long)blockIdx.x * 256 + threadIdx.x;
    if (i >= n4) return;
    const float inv = 1.0f / rs[i >> 5];
    float4 v = ((float4*)y)[i];
    v.x *= inv; v.y *= inv; v.z *= inv; v.w *= inv;
    ((float4*)y)[i] = v;
}

extern "C" void kernel_launch(void* const* d_in, const int* in_sizes, int n_in,
                              void* d_out, int out_size, void* d_ws, size_t ws_size,
                              hipStream_t stream)
{
    (void)n_in; (void)out_size; (void)ws_size;
    const int N = in_sizes[0] / D_;   // 100000 vertices
    const int E = in_sizes[1] / D_;   // 50000 hyperedges
    const int M = in_sizes[2];        // 1000000 incidences

    const float* v            = (const float*)d_in[0];
    const float* e            = (const float*)d_in[1];
    const int*   vidx         = (const int*)d_in[2];
    const int*   eidx         = (const int*)d_in[3];
    const float* v_reg_weight = (const float*)d_in[4];
    const float* e_reg_weight = (const float*)d_in[5];
    const float* v_reg_sum    = (const float*)d_in[6];
    const float* e_reg_sum    = (const float*)d_in[7];
    const float* W_v2e        = (const float*)d_in[8];
    const float* W_e2v        = (const float*)d_in[9];
    const float* b_v          = (const float*)d_in[10];
    const float* b_e          = (const float*)d_in[11];
    const float* v_weight     = (const float*)d_in[12];

    float* v_out = (float*)d_out;                       // [N,128]
    float* e_out = (float*)d_out + (size_t)N * D_;      // [E,128]
    float* msg   = (float*)d_ws;                        // N*128 fp32, reused

    const dim3 blk(256);
    const long v4 = (long)N * D_ / 4;
    const long e4 = (long)E * D_ / 4;
    const unsigned scat_blocks = (unsigned)((M * 32 + 255) / 256);

    // 1) msg = relu(v @ W_v2e + b_v) * v_weight          (WMMA f32)
    gemm_relu_rowscale<true><<<(N + TM - 1) / TM, blk, 0, stream>>>(
        v, W_v2e, b_v, v_weight, 1.0f, msg, N);
    // 2) e_out = e
    rowscale_copy<<<(unsigned)((e4 + 255) / 256), blk, 0, stream>>>(
        e, nullptr, 1.0f, e_out, e4);
    // 3) e_out[eidx[m]] += msg[vidx[m]] * v_reg_weight[m]
    scatter_rows_atomic<<<scat_blocks, blk, 0, stream>>>(
        msg, vidx, eidx, v_reg_weight, e_out, M);
    // 4) e_out /= e_reg_sum   (e_out final)
    rowdiv_inplace<<<(unsigned)((e4 + 255) / 256), blk, 0, stream>>>(
        e_out, e_reg_sum, e4);
    // 5) msg = relu(e_out @ W_e2v + b_e) / 3             (WMMA f32)
    gemm_relu_rowscale<false><<<(E + TM - 1) / TM, blk, 0, stream>>>(
        e_out, W_e2v, b_e, nullptr, 1.0f / 3.0f, msg, E);
    // 6) v_out = v * v_weight * 4
    rowscale_copy<<<(unsigned)((v4 + 255) / 256), blk, 0, stream>>>(
        v, v_weight, 4.0f, v_out, v4);
    // 7) v_out[vidx[m]] += msg[eidx[m]] * e_reg_weight[m]
    scatter_rows_atomic<<<scat_blocks, blk, 0, stream>>>(
        msg, eidx, vidx, e_reg_weight, v_out, M);
    // 8) v_out /= v_reg_sum   (v_out final)
    rowdiv_inplace<<<(unsigned)((v4 + 255) / 256), blk, 0, stream>>>(
        v_out, v_reg_sum, v4);
}